// HEAD_12111807775050
// MI455X (gfx1250) — compile-verified
//
#include <hip/hip_runtime.h>
#include <math.h>

// ---------------------------------------------------------------------------
// YOLOv3 head decode for MI455X (gfx1250).
// Memory-bound streaming op (~248 MB moved, ~10.6 us floor @ 23.3 TB/s).
// Strategy: stage 192 rows x 85ch per block into LDS via coalesced 16B
// async global->LDS transfers (CDNA5 GLOBAL_LOAD_ASYNC_TO_LDS_B128 path,
// tracked by ASYNCcnt), compute per-row decode from LDS (bank-conflict-free:
// stride 85 words -> 21*lane mod 64 is a bijection over wave32), stage the
// 6-float results in LDS, flush with coalesced non-temporal b128 stores.
// ---------------------------------------------------------------------------

#ifndef __has_builtin
#define __has_builtin(x) 0
#endif

#if defined(__AMDGCN__) && \
    __has_builtin(__builtin_amdgcn_global_load_async_to_lds_b128) && \
    __has_builtin(__builtin_amdgcn_s_wait_asynccnt)
#define USE_ASYNC_LDS 1
#else
#define USE_ASYNC_LDS 0
#endif

// native vector types (HIP's float4 is a class; builtins need real vectors)
typedef float v4f __attribute__((ext_vector_type(4)));
typedef int   v4i __attribute__((ext_vector_type(4)));
typedef __attribute__((address_space(1))) v4i* gv4i_p;  // global
typedef __attribute__((address_space(3))) v4i* lv4i_p;  // LDS

#define ROWS_PER_BLOCK 192   // divides 32448 / 129792 / 519168 exactly
#define NCH            85
#define CONF_TH        0.5f
#define IMG_SIZE       416.0f

#define N_TOTAL  681408
#define BASE_M   32448       // 104*104*3
#define BASE_L   162240      // + 208*208*3

__global__ __launch_bounds__(ROWS_PER_BLOCK)
void yolo_decode_kernel(const float* __restrict__ p_small,
                        const float* __restrict__ p_middle,
                        const float* __restrict__ p_large,
                        const float* __restrict__ pre_scale,
                        float* __restrict__ out)
{
    __shared__ float sIn[ROWS_PER_BLOCK * NCH];  // 16320 floats = 65280 B
    __shared__ float sOut[ROWS_PER_BLOCK * 6];   //  1152 floats =  4608 B

    const int tid = threadIdx.x;
    const int r0  = blockIdx.x * ROWS_PER_BLOCK; // global row base; blocks never straddle levels

    const float* src;
    int S, base, ds;
    if (r0 < BASE_M)      { src = p_small;  S = 104; base = 0;      ds = 3; }
    else if (r0 < BASE_L) { src = p_middle; S = 208; base = BASE_M; ds = 4; }
    else                  { src = p_large;  S = 416; base = BASE_L; ds = 5; }

    // ---- stage 1: coalesced 16B copy of 192 rows into LDS -----------------
    // block byte base = (r0-base)*340, a multiple of 192*340=65280 -> 16B aligned
    const v4f* gsrc = (const v4f*)(src + (size_t)(r0 - base) * NCH);
    const int nvec = ROWS_PER_BLOCK * NCH / 4;   // 4080 x 16B per block
    for (int i = tid; i < nvec; i += ROWS_PER_BLOCK) {
#if USE_ASYNC_LDS
        __builtin_amdgcn_global_load_async_to_lds_b128(
            (gv4i_p)(gsrc + i),
            (lv4i_p)(&sIn[i * 4]),
            0, 0);
#else
        ((v4f*)sIn)[i] = __builtin_nontemporal_load(gsrc + i);
#endif
    }
#if USE_ASYNC_LDS
    __builtin_amdgcn_s_wait_asynccnt(0);   // s_wait_asynccnt 0
#endif
    __syncthreads();

    // ---- stage 2: per-row decode (one thread = one row) -------------------
    const float inv_img = 1.0f / IMG_SIZE;
    const float aw0 = pre_scale[ds * 6 + 0] * inv_img;
    const float ah0 = pre_scale[ds * 6 + 1] * inv_img;
    const float aw1 = pre_scale[ds * 6 + 2] * inv_img;
    const float ah1 = pre_scale[ds * 6 + 3] * inv_img;
    const float aw2 = pre_scale[ds * 6 + 4] * inv_img;
    const float ah2 = pre_scale[ds * 6 + 5] * inv_img;

    {
        const float* row = &sIn[tid * NCH];       // LDS, conflict-free stride
        const int r  = r0 + tid;                  // global row
        const int ir = r - base;                  // row within this level
        const int a  = ir % 3;                    // anchor index
        const int i1 = (ir / 3) % S;              // y grid offset (axis 1)
        const int i0 = ir / (3 * S);              // x grid offset (axis 0)
        const float invS = 1.0f / (float)S;

        const float conf = row[0];
        const float x = (1.0f / (1.0f + __expf(-row[1])) + (float)i0) * invS;
        const float y = (1.0f / (1.0f + __expf(-row[2])) + (float)i1) * invS;
        const float aw = (a == 0) ? aw0 : ((a == 1) ? aw1 : aw2);
        const float ah = (a == 0) ? ah0 : ((a == 1) ? ah1 : ah2);
        const float w = __expf(row[3]) * aw;
        const float h = __expf(row[4]) * ah;

        // first-occurrence argmax over 80 class logits (matches jnp.argmax)
        float best = row[5];
        int   bi   = 0;
        #pragma unroll 8
        for (int c = 1; c < 80; ++c) {
            const float v = row[5 + c];
            if (v > best) { best = v; bi = c; }
        }

        const float m = (conf > CONF_TH) ? 1.0f : 0.0f;
        float* o = &sOut[tid * 6];
        o[0] = x * m;
        o[1] = y * m;
        o[2] = w * m;
        o[3] = h * m;
        o[4] = (float)bi * m;
        o[5] = conf * m;
    }
    __syncthreads();

    // ---- stage 3: coalesced non-temporal flush of results -----------------
    // out byte base = r0*24, multiple of 192*24=4608 -> 16B aligned
    v4f* gout = (v4f*)(out + (size_t)r0 * 6);
    const int novec = ROWS_PER_BLOCK * 6 / 4;    // 288 x 16B per block
    for (int i = tid; i < novec; i += ROWS_PER_BLOCK) {
        const v4f val = ((const v4f*)sOut)[i];
        __builtin_nontemporal_store(val, gout + i);
    }
}

extern "C" void kernel_launch(void* const* d_in, const int* in_sizes, int n_in,
                              void* d_out, int out_size, void* d_ws, size_t ws_size,
                              hipStream_t stream) {
    (void)in_sizes; (void)n_in; (void)d_ws; (void)ws_size; (void)out_size;
    const float* p_small   = (const float*)d_in[0];
    const float* p_middle  = (const float*)d_in[1];
    const float* p_large   = (const float*)d_in[2];
    const float* pre_scale = (const float*)d_in[3];
    float* out = (float*)d_out;

    const int blocks = N_TOTAL / ROWS_PER_BLOCK;  // 3549, exact
    hipLaunchKernelGGL(yolo_decode_kernel, dim3(blocks), dim3(ROWS_PER_BLOCK),
                       0, stream, p_small, p_middle, p_large, pre_scale, out);
}